// GNN_node_clasif_model_28793460752815
// MI455X (gfx1250) — compile-verified
//
#include <hip/hip_runtime.h>
#include <cmath>

typedef __attribute__((ext_vector_type(2))) float v2f;
typedef __attribute__((ext_vector_type(8))) float v8f;

#define NNODES 100000
#define NEDGES 3200000
#define FDIM1 128
#define FDIM2 64
#define NCLS  40

// ---------------------------------------------------------------------------
// 1) degree init (self loop contributes 1)
__global__ void k_init_deg(float* __restrict__ deg) {
    int i = blockIdx.x * 256 + threadIdx.x;
    if (i < NNODES) deg[i] = 1.0f;
}

// 2) in-degree accumulation over edges
__global__ void k_degree(const int* __restrict__ ei, float* __restrict__ deg) {
    int e = blockIdx.x * 256 + threadIdx.x;
    if (e < NEDGES) {
        int dst = ei[NEDGES + e];
        unsafeAtomicAdd(&deg[dst], 1.0f);
    }
}

// 3) deg -> rsqrt(deg)   (deg >= 1 always due to self loop)
__global__ void k_rsqrt(float* __restrict__ deg) {
    int i = blockIdx.x * 256 + threadIdx.x;
    if (i < NNODES) deg[i] = rsqrtf(deg[i]);
}

// ---------------------------------------------------------------------------
// 4) fp32 WMMA GEMM: C[M x N] = A[M x K] * B[K x N]
//    one wave per 16x16 output tile; M%16==0, N%16==0, K%4==0; grid sized
//    exactly so EXEC is all-1s for every wmma (ISA requirement).
//    A-frag (16x4 f32): lanes 0-15 hold K={k0,k0+1}, lanes 16-31 K={k0+2,k0+3}
//    B-frag (4x16 f32): same K split across lane halves, N = lane%16
//    C/D: VGPR j = rows {j, j+8} of the tile.
__global__ void k_gemm_wmma(const float* __restrict__ A,
                            const float* __restrict__ B,
                            float* __restrict__ C,
                            int K, int N) {
    int wave = (blockIdx.x * blockDim.x + threadIdx.x) >> 5;
    int lane = threadIdx.x & 31;
    int ntiles = N >> 4;
    int mt = wave / ntiles;
    int nt = wave - mt * ntiles;
    int half = lane >> 4;      // 0: lower K pair, 1: upper K pair
    int lr   = lane & 15;

    const float* arow = A + (long)(mt * 16 + lr) * K + 2 * half;
    const float* bcol = B + (long)(2 * half) * N + nt * 16 + lr;

    v8f acc = {};
    for (int k0 = 0; k0 < K; k0 += 4) {
        v2f a = *(const v2f*)(arow + k0);          // global_load_b64
        v2f b;
        b.x = bcol[(long)k0 * N];
        b.y = bcol[(long)k0 * N + N];
        acc = __builtin_amdgcn_wmma_f32_16x16x4_f32(
            /*neg_a=*/false, a, /*neg_b=*/false, b,
            /*c_mod=*/(short)0, acc, /*reuse_a=*/false, /*reuse_b=*/false);
    }

    float* crow = C + (long)(mt * 16 + 8 * half) * N + nt * 16 + lr;
#pragma unroll
    for (int j = 0; j < 8; ++j) crow[(long)j * N] = acc[j];
}

// ---------------------------------------------------------------------------
// 5) initialize aggregation buffer with the self-loop term:
//    agg[i][f] = h[i][f] * dinv[i]^2
__global__ void k_selfloop(const float* __restrict__ h,
                           const float* __restrict__ dinv,
                           float* __restrict__ agg) {
    int idx = blockIdx.x * 256 + threadIdx.x;
    if (idx < NNODES * FDIM2) {
        int i = idx >> 6;
        float d = dinv[i];
        agg[idx] = h[idx] * d * d;
    }
}

// 6) edge scatter: one wave32 per edge, 2 features per lane.
//    coalesced 256B gather of h[src], contiguous f32 atomic adds into agg[dst].
__global__ void k_scatter(const int* __restrict__ ei,
                          const float* __restrict__ dinv,
                          const float* __restrict__ h,
                          float* __restrict__ agg) {
    int wave = (blockIdx.x * blockDim.x + threadIdx.x) >> 5;
    int lane = threadIdx.x & 31;
    if (wave >= NEDGES) return;
    int src = ei[wave];
    int dst = ei[NEDGES + wave];
    float w = dinv[src] * dinv[dst];
    const float* hs = h + (long)src * FDIM2;
    float* ad = agg + (long)dst * FDIM2;
    unsafeAtomicAdd(&ad[lane],      hs[lane]      * w);
    unsafeAtomicAdd(&ad[lane + 32], hs[lane + 32] * w);
}

// 7) bias + ReLU
__global__ void k_bias_relu(const float* __restrict__ agg,
                            const float* __restrict__ b,
                            float* __restrict__ act) {
    int idx = blockIdx.x * 256 + threadIdx.x;
    if (idx < NNODES * FDIM2) {
        int f = idx & 63;
        act[idx] = fmaxf(agg[idx] + b[f], 0.0f);
    }
}

// ---------------------------------------------------------------------------
// 8) fused: relu(agg2 + b2) @ Wf + bf, then log_softmax over 40 classes.
//    One thread per node; Wf (10KB) and bf live in cache.
__global__ void k_final(const float* __restrict__ agg2,
                        const float* __restrict__ b2,
                        const float* __restrict__ Wf,
                        const float* __restrict__ bf,
                        float* __restrict__ out) {
    int i = blockIdx.x * 256 + threadIdx.x;
    if (i >= NNODES) return;
    float logits[NCLS];
#pragma unroll
    for (int j = 0; j < NCLS; ++j) logits[j] = bf[j];
    const float* a = agg2 + (long)i * FDIM2;
    for (int f = 0; f < FDIM2; ++f) {
        float v = fmaxf(a[f] + b2[f], 0.0f);
        const float* wr = Wf + f * NCLS;
#pragma unroll
        for (int j = 0; j < NCLS; ++j) logits[j] = fmaf(v, wr[j], logits[j]);
    }
    float m = logits[0];
#pragma unroll
    for (int j = 1; j < NCLS; ++j) m = fmaxf(m, logits[j]);
    float s = 0.0f;
#pragma unroll
    for (int j = 0; j < NCLS; ++j) s += expf(logits[j] - m);
    float lse = m + logf(s);
    float* o = out + (long)i * NCLS;
#pragma unroll
    for (int j = 0; j < NCLS; ++j) o[j] = logits[j] - lse;
}

// ---------------------------------------------------------------------------
extern "C" void kernel_launch(void* const* d_in, const int* in_sizes, int n_in,
                              void* d_out, int out_size, void* d_ws, size_t ws_size,
                              hipStream_t stream) {
    const float* x  = (const float*)d_in[0];
    const int*   ei = (const int*)d_in[1];   // [2, E] src row then dst row
    const float* W1 = (const float*)d_in[2];
    const float* b1 = (const float*)d_in[3];
    const float* W2 = (const float*)d_in[4];
    const float* b2 = (const float*)d_in[5];
    const float* Wf = (const float*)d_in[6];
    const float* bf = (const float*)d_in[7];
    float* out = (float*)d_out;

    // workspace layout (floats)
    float* ws   = (float*)d_ws;
    float* dinv = ws;                    // NNODES         (deg -> rsqrt in place)
    float* bufA = ws + 131072;           // NNODES*64
    float* bufB = bufA + NNODES * FDIM2; // NNODES*64
    float* bufC = bufB + NNODES * FDIM2; // NNODES*64

    const int nodeBlk = (NNODES + 255) / 256;            // 391
    const int featBlk = (NNODES * FDIM2 + 255) / 256;    // 25000
    const int edgeThreadBlk = (NEDGES + 255) / 256;      // 12500
    const int edgeWaveBlk = NEDGES / 8;                  // 400000 (8 waves/block)
    const int gemmBlk = (NNODES / 16) * (FDIM2 / 16) / 8; // 3125 (8 waves/block)

    // graph normalization
    k_init_deg<<<nodeBlk, 256, 0, stream>>>(dinv);
    k_degree<<<edgeThreadBlk, 256, 0, stream>>>(ei, dinv);
    k_rsqrt<<<nodeBlk, 256, 0, stream>>>(dinv);

    // layer 1: h1 = x @ W1 ; agg1 = scatter ; act1 = relu(agg1 + b1)
    k_gemm_wmma<<<gemmBlk, 256, 0, stream>>>(x, W1, bufA, FDIM1, FDIM2);
    k_selfloop<<<featBlk, 256, 0, stream>>>(bufA, dinv, bufB);
    k_scatter<<<edgeWaveBlk, 256, 0, stream>>>(ei, dinv, bufA, bufB);
    k_bias_relu<<<featBlk, 256, 0, stream>>>(bufB, b1, bufA);

    // layer 2: h2 = act1 @ W2 ; agg2 = scatter
    k_gemm_wmma<<<gemmBlk, 256, 0, stream>>>(bufA, W2, bufC, FDIM2, FDIM2);
    k_selfloop<<<featBlk, 256, 0, stream>>>(bufC, dinv, bufB);
    k_scatter<<<edgeWaveBlk, 256, 0, stream>>>(ei, dinv, bufC, bufB);

    // head: relu + linear + log_softmax
    k_final<<<nodeBlk, 256, 0, stream>>>(bufB, b2, Wf, bf, out);
}